// SphericalBasisLayer_84026740179773
// MI455X (gfx1250) — compile-verified
//
#include <hip/hip_runtime.h>

#ifndef M_PI
#define M_PI 3.14159265358979323846
#endif

#define NL 7          // NUM_SPHERICAL
#define NK 6          // NUM_RADIAL
#define LK 42         // NL*NK
#define TILE 16       // pairs per wave tile (WMMA M)
#define CBF_STRIDE 49 // LDS row stride (pad vs bank conflicts)

typedef __attribute__((ext_vector_type(2))) float v2f;
typedef __attribute__((ext_vector_type(8))) float v8f;

#if defined(__has_builtin)
#  if __has_builtin(__builtin_amdgcn_wmma_f32_16x16x4_f32)
#    define HAVE_WMMA4 1
#  else
#    define HAVE_WMMA4 0
#  endif
#else
#  define HAVE_WMMA4 0
#endif

// spherical Bessel j_l(t) via upward recurrence (matches reference _sph_jn)
__device__ __forceinline__ float sph_jl(float t, int l) {
  float s, c;
  sincosf(t, &s, &c);
  float invt = 1.0f / t;
  float j0 = s * invt;
  if (l == 0) return j0;
  float j1 = (s * invt - c) * invt;   // sin/t^2 - cos/t
  for (int i = 1; i < l; ++i) {
    float jn = (float)(2 * i + 1) * invt * j1 - j0;
    j0 = j1; j1 = jn;
  }
  return j1;
}

// DimeNet envelope, p = 6: 1/x - 28 x^5 + 48 x^6 - 21 x^7
__device__ __forceinline__ float envelope6(float x) {
  float x2 = x * x;
  float x5 = x2 * x2 * x;
  return 1.0f / x + x5 * (-28.0f + x * (48.0f - 21.0f * x));
}

// ---------------------------------------------------------------------------
// Init kernel (1 wave): Bessel zeros Z (interlaced bisection), norms,
// and monomial-basis Legendre coefficient matrix G[8][48] (incl Y_l^0 prefactor)
// ws layout (floats): [0..41]=Z, [42..83]=NORM, [84..467]=G, pad to 512.
// ---------------------------------------------------------------------------
__global__ void sbl_init_kernel(float* __restrict__ ws) {
  const int lane = threadIdx.x;
  __shared__ float prev[NL + NK];
  if (lane < NL + NK) prev[lane] = (float)(lane + 1) * 3.14159265358979323846f;
  __syncthreads();
  if (lane < NK) ws[lane] = prev[lane];           // zeros of j_0 = n*pi
  for (int l = 1; l < NL; ++l) {
    int cnt = NL + NK - l;
    float lo = 1.0f, hi = 2.0f, flo = 1.0f;
    if (lane < cnt) { lo = prev[lane]; hi = prev[lane + 1]; flo = sph_jl(lo, l); }
    for (int it = 0; it < 48; ++it) {
      float mid = 0.5f * (lo + hi);
      float fm = sph_jl(mid, l);
      bool same = (fm > 0.0f) == (flo > 0.0f);
      lo  = same ? mid : lo;
      flo = same ? fm  : flo;
      hi  = same ? hi  : mid;
    }
    __syncthreads();
    if (lane < cnt) prev[lane] = 0.5f * (lo + hi);
    __syncthreads();
    if (lane < NK) ws[l * NK + lane] = prev[lane];
  }
  __syncthreads();
  for (int idx = lane; idx < LK; idx += 32) {      // NORM = 1/sqrt(0.5*j_{l+1}(z)^2)
    int l = idx / NK;
    float z = ws[idx];
    float jn = sph_jl(z, l + 1);
    ws[LK + idx] = rsqrtf(0.5f * jn * jn);
  }
  if (lane == 0) {
    float* G = ws + 2 * LK;
    for (int i = 0; i < 8 * 48; ++i) G[i] = 0.0f;
    float sphc[NL];
    for (int l = 0; l < NL; ++l)
      sphc[l] = sqrtf((float)(2 * l + 1) / (4.0f * (float)M_PI));
    auto SET = [&](int k, int l, float v) {
      for (int n = 0; n < NK; ++n) G[k * 48 + l * NK + n] = v * sphc[l];
    };
    // Legendre polynomials in monomial basis
    SET(0, 0,  1.0f);
    SET(1, 1,  1.0f);
    SET(0, 2, -0.5f);    SET(2, 2,  1.5f);
    SET(1, 3, -1.5f);    SET(3, 3,  2.5f);
    SET(0, 4,  0.375f);  SET(2, 4, -3.75f);   SET(4, 4,  4.375f);
    SET(1, 5,  1.875f);  SET(3, 5, -8.75f);   SET(5, 5,  7.875f);
    SET(0, 6, -0.3125f); SET(2, 6,  6.5625f); SET(4, 6, -19.6875f); SET(6, 6, 14.4375f);
  }
}

// ---------------------------------------------------------------------------
// Edge kernel: per edge compute normalized direction (+x in .w) and the full
// 42-entry radial Bessel basis (with envelope) once.
// ---------------------------------------------------------------------------
__global__ __launch_bounds__(256) void sbl_edge_kernel(
    const float* __restrict__ dvec, const float* __restrict__ wsc,
    float4* __restrict__ nd, float* __restrict__ rbf, int E) {
  __shared__ float sZ[LK], sN[LK];
  for (int i = threadIdx.x; i < LK; i += blockDim.x) { sZ[i] = wsc[i]; sN[i] = wsc[LK + i]; }
  __syncthreads();
  int e = blockIdx.x * blockDim.x + threadIdx.x;
  if (e >= E) return;
  float dx = dvec[3 * e + 0], dy = dvec[3 * e + 1], dz = dvec[3 * e + 2];
  float dist = sqrtf(dx * dx + dy * dy + dz * dz);
  float inv = 1.0f / dist;
  float x = dist * 0.2f;                           // dist / CUTOFF
  nd[e] = make_float4(dx * inv, dy * inv, dz * inv, x);
  float env = envelope6(x);
  float* __restrict__ row = rbf + (size_t)e * LK;
  for (int j = 0; j < LK; ++j) {
    int l = j / NK;
    row[j] = sN[j] * sph_jl(sZ[j] * x, l) * env;
  }
}

// ---------------------------------------------------------------------------
// Pair kernel: one wave per 16-pair tile (each wave handles ~4 tiles).
//  Stage A: lanes 0-15 gather dirs, compute cos + sender idx (broadcast later
//           via __shfl, no LDS round trip).
//  Stage B: cbf_expanded(16x48) = powers(t)(16x8) @ G(8x48) via two
//           V_WMMA_F32_16X16X4_F32 per 16-col tile (3 tiles); B operand is
//           tile-invariant -> built branch-free and hoisted to registers.
//  Stage C: coalesced fuse: out[p, j] = rbf[s_p, j] * cbf_lds[p, j], NT stores.
// ---------------------------------------------------------------------------
template <int USE_WS>
__global__ __launch_bounds__(256) void sbl_pair_kernel(
    const float* __restrict__ dvec,
    const int* __restrict__ snd, const int* __restrict__ rcv,
    const float* __restrict__ wsc,
    const float4* __restrict__ nd, const float* __restrict__ rbf,
    float* __restrict__ out, int ntiles) {
  __shared__ float sZ[LK], sN[LK];
  __shared__ float sG[8 * 48];
  __shared__ float sCbf[8][16 * CBF_STRIDE];
  if (!USE_WS) {
    for (int i = threadIdx.x; i < LK; i += blockDim.x) { sZ[i] = wsc[i]; sN[i] = wsc[LK + i]; }
  }
  for (int i = threadIdx.x; i < 8 * 48; i += blockDim.x) sG[i] = wsc[2 * LK + i];
  __syncthreads();

  const int lane = threadIdx.x & 31;
  const int wv = threadIdx.x >> 5;
  const int m = lane & 15;       // row (pair) for A, column for B/D
  const int hig = lane >> 4;     // lane group
  int wave = blockIdx.x * (blockDim.x >> 5) + wv;
  int nwaves = gridDim.x * (blockDim.x >> 5);

  // Tile-invariant B operand (coefficient matrix G), branch-free per ISA
  // B 4x16 f32 layout: lanes0-15 rows {K0,K1}/{K4,K5}; lanes16-31 {K2,K3}/{K6,K7}
  v2f B0[3], B1[3];
#pragma unroll
  for (int tc = 0; tc < 3; ++tc) {
    int n = m + tc * 16;
    B0[tc].x = sG[(2 * hig + 0) * 48 + n];
    B0[tc].y = sG[(2 * hig + 1) * 48 + n];
    B1[tc].x = sG[(2 * hig + 4) * 48 + n];
    B1[tc].y = sG[(2 * hig + 5) * 48 + n];
  }

  for (int tile = wave; tile < ntiles; tile += nwaves) {
    const long long p0 = (long long)tile * TILE;
    float cosv = 0.0f;
    int sv = 0;
    if (lane < 16) {
      long long p = p0 + lane;
      sv = snd[p];
      int r = rcv[p];
      float ax, ay, az, bx, by, bz;
      if (USE_WS) {
        float4 a = nd[sv]; ax = a.x; ay = a.y; az = a.z;
        float4 b = nd[r];  bx = b.x; by = b.y; bz = b.z;
        __builtin_prefetch(rbf + (size_t)sv * LK, 0, 1);       // first 128B line
        __builtin_prefetch(rbf + (size_t)sv * LK + 32, 0, 1);  // rest of 168B row
      } else {
        float d0 = dvec[3 * sv], d1 = dvec[3 * sv + 1], d2 = dvec[3 * sv + 2];
        float ia = 1.0f / sqrtf(d0 * d0 + d1 * d1 + d2 * d2);
        ax = d0 * ia; ay = d1 * ia; az = d2 * ia;
        float u0 = dvec[3 * r], u1 = dvec[3 * r + 1], u2 = dvec[3 * r + 2];
        float ib = 1.0f / sqrtf(u0 * u0 + u1 * u1 + u2 * u2);
        bx = u0 * ib; by = u1 * ib; bz = u2 * ib;
      }
      cosv = ax * bx + ay * by + az * bz;
    }
    // cross-lane broadcast of cos (ds_bpermute, no LDS memory / DScnt)
    float t = __shfl(cosv, m, 32);
    float t2 = t * t, t3 = t2 * t;
    // A 16x4 f32 layout: lanes0-15 hold {K0,K1}={1,t}; lanes16-31 {K2,K3}={t2,t3}
    v2f a0, a1;
    a0.x = hig ? t2      : 1.0f;
    a0.y = hig ? t3      : t;
    a1.x = hig ? t3 * t3 : t2 * t2;
    a1.y = hig ? 0.0f    : t2 * t3;
#if HAVE_WMMA4
#pragma unroll
    for (int tc = 0; tc < 3; ++tc) {
      int n = m + tc * 16;
      v8f acc = {};
      acc = __builtin_amdgcn_wmma_f32_16x16x4_f32(false, a0, false, B0[tc], (short)0, acc, false, false);
      acc = __builtin_amdgcn_wmma_f32_16x16x4_f32(false, a1, false, B1[tc], (short)0, acc, false, false);
      // D layout: VGPR r -> row (r + 8*laneGroup), col = lane%16 (+16*tc)
#pragma unroll
      for (int r8 = 0; r8 < 8; ++r8)
        sCbf[wv][(r8 + 8 * hig) * CBF_STRIDE + n] = acc[r8];
    }
#else
    if (lane < 16) {   // VALU fallback: direct Legendre recurrence
      float vals[NL];
      vals[0] = 1.0f; vals[1] = t;
      float Pp = 1.0f, Pc = t;
      for (int l = 2; l < NL; ++l) {
        float Pn = ((float)(2 * l - 1) * t * Pc - (float)(l - 1) * Pp) / (float)l;
        Pp = Pc; Pc = Pn; vals[l] = Pn;
      }
      for (int l = 0; l < NL; ++l) {
        float c = sqrtf((float)(2 * l + 1) / (4.0f * (float)M_PI)) * vals[l];
        for (int n = 0; n < NK; ++n) sCbf[wv][lane * CBF_STRIDE + l * NK + n] = c;
      }
    }
#endif
    // Stage C: coalesced fused write-out (16 rows x 42 cols)
    for (int mm = 0; mm < TILE; ++mm) {
      int s = __shfl(sv, mm, 32);
      const size_t rbase = (size_t)s * LK;
      float xs = 0.0f, envs = 0.0f;
      if (!USE_WS) {
        float d0 = dvec[3 * s], d1 = dvec[3 * s + 1], d2 = dvec[3 * s + 2];
        float dist = sqrtf(d0 * d0 + d1 * d1 + d2 * d2);
        xs = dist * 0.2f;
        envs = envelope6(xs);
      }
      float* __restrict__ orow = out + (size_t)(p0 + mm) * LK;
      int j = lane;
      float rv;
      if (USE_WS) rv = rbf[rbase + j];
      else        rv = sN[j] * sph_jl(sZ[j] * xs, j / NK) * envs;
      __builtin_nontemporal_store(rv * sCbf[wv][mm * CBF_STRIDE + j], orow + j);
      if (lane < LK - 32) {
        int j2 = lane + 32;
        float rv2;
        if (USE_WS) rv2 = rbf[rbase + j2];
        else        rv2 = sN[j2] * sph_jl(sZ[j2] * xs, j2 / NK) * envs;
        __builtin_nontemporal_store(rv2 * sCbf[wv][mm * CBF_STRIDE + j2], orow + j2);
      }
    }
  }
}

// ---------------------------------------------------------------------------
// Tail kernel: one thread per output element for P % 16 leftover pairs.
// ---------------------------------------------------------------------------
template <int USE_WS>
__global__ void sbl_tail_kernel(const float* __restrict__ dvec,
                                const int* __restrict__ snd, const int* __restrict__ rcv,
                                const float* __restrict__ wsc,
                                const float4* __restrict__ nd, const float* __restrict__ rbf,
                                float* __restrict__ out, long long p_lo, long long P) {
  long long tix = (long long)blockIdx.x * blockDim.x + threadIdx.x;
  long long nelem = (P - p_lo) * LK;
  if (tix >= nelem) return;
  long long p = p_lo + tix / LK;
  int j = (int)(tix % LK);
  int s = snd[p], r = rcv[p];
  float ax, ay, az, bx, by, bz, xs = 0.0f;
  if (USE_WS) {
    float4 a = nd[s]; ax = a.x; ay = a.y; az = a.z;
    float4 b = nd[r]; bx = b.x; by = b.y; bz = b.z;
  } else {
    float d0 = dvec[3 * s], d1 = dvec[3 * s + 1], d2 = dvec[3 * s + 2];
    float dist = sqrtf(d0 * d0 + d1 * d1 + d2 * d2);
    float ia = 1.0f / dist;
    ax = d0 * ia; ay = d1 * ia; az = d2 * ia;
    xs = dist * 0.2f;
    float u0 = dvec[3 * r], u1 = dvec[3 * r + 1], u2 = dvec[3 * r + 2];
    float ib = 1.0f / sqrtf(u0 * u0 + u1 * u1 + u2 * u2);
    bx = u0 * ib; by = u1 * ib; bz = u2 * ib;
  }
  float cv = ax * bx + ay * by + az * bz;
  int l = j / NK;
  float Pl = 1.0f;
  if (l >= 1) {
    float Pp = 1.0f, Pc = cv;
    for (int i = 1; i < l; ++i) {
      float Pn = ((float)(2 * i + 1) * cv * Pc - (float)i * Pp) / (float)(i + 1);
      Pp = Pc; Pc = Pn;
    }
    Pl = Pc;
  }
  float cb = sqrtf((float)(2 * l + 1) / (4.0f * (float)M_PI)) * Pl;
  float rv;
  if (USE_WS) rv = rbf[(size_t)s * LK + j];
  else        rv = wsc[LK + j] * sph_jl(wsc[j] * xs, l) * envelope6(xs);
  out[(size_t)p * LK + j] = rv * cb;
}

extern "C" void kernel_launch(void* const* d_in, const int* in_sizes, int n_in,
                              void* d_out, int out_size, void* d_ws, size_t ws_size,
                              hipStream_t stream) {
  const float* dvec = (const float*)d_in[0];
  const int* snd = (const int*)d_in[1];
  const int* rcv = (const int*)d_in[2];
  float* out = (float*)d_out;
  const int E = in_sizes[0] / 3;
  const long long P = in_sizes[1];
  float* ws = (float*)d_ws;

  const size_t CONST_F = 512;                       // Z|NORM|G constants region
  float4* nd = (float4*)(ws + CONST_F);
  float* rbf = ws + CONST_F + (size_t)E * 4;
  const size_t need = (CONST_F + (size_t)E * 4 + (size_t)E * LK) * sizeof(float);
  const bool use_ws = (ws_size >= need);

  sbl_init_kernel<<<1, 32, 0, stream>>>(ws);

  const int ntiles = (int)(P / TILE);
  const long long p_tail = (long long)ntiles * TILE;
  // ~4 tiles per wave: amortizes staging/B-setup, still >30k waves in flight
  const int blocks = ntiles > 0 ? (ntiles + 31) / 32 : 1;

  if (use_ws) {
    sbl_edge_kernel<<<(E + 255) / 256, 256, 0, stream>>>(dvec, ws, nd, rbf, E);
    if (ntiles > 0)
      sbl_pair_kernel<1><<<blocks, 256, 0, stream>>>(dvec, snd, rcv, ws, nd, rbf, out, ntiles);
    if (p_tail < P) {
      long long nel = (P - p_tail) * LK;
      sbl_tail_kernel<1><<<(int)((nel + 255) / 256), 256, 0, stream>>>(
          dvec, snd, rcv, ws, nd, rbf, out, p_tail, P);
    }
  } else {
    if (ntiles > 0)
      sbl_pair_kernel<0><<<blocks, 256, 0, stream>>>(dvec, snd, rcv, ws, nullptr, nullptr, out, ntiles);
    if (p_tail < P) {
      long long nel = (P - p_tail) * LK;
      sbl_tail_kernel<0><<<(int)((nel + 255) / 256), 256, 0, stream>>>(
          dvec, snd, rcv, ws, nullptr, nullptr, out, p_tail, P);
    }
  }
}